// NodeUpdateLayer_15040975470646
// MI455X (gfx1250) — compile-verified
//
#include <hip/hip_runtime.h>
#include <hip/hip_bf16.h>

#define N_NODES  100000
#define N_EDGES  640000
#define NODE_DIM 128
#define EDGE_DIM 64
#define EPS      1e-6f

typedef float v2f __attribute__((ext_vector_type(2)));
typedef float v8f __attribute__((ext_vector_type(8)));

// Monotonic f32 <-> u32 encoding so atomicMax(u32) == float max (exact, total order).
__device__ __forceinline__ unsigned enc_f32(float f) {
  unsigned b = __float_as_uint(f);
  return (b & 0x80000000u) ? ~b : (b | 0x80000000u);
}
__device__ __forceinline__ float dec_f32(unsigned k) {
  unsigned b = (k & 0x80000000u) ? (k & 0x7FFFFFFFu) : ~k;
  return __uint_as_float(b);
}

// ---------------- init: zero the accumulators (must run every launch) ------
__global__ void k_init(float* __restrict__ messages, float* __restrict__ sum_exp,
                       unsigned* __restrict__ max_enc) {
  size_t i = (size_t)blockIdx.x * blockDim.x + threadIdx.x;
  if (i < (size_t)N_NODES * NODE_DIM) messages[i] = 0.f;
  if (i < N_NODES) { sum_exp[i] = 0.f; max_enc[i] = 0u; }
}

// ---------------- edge scores: one wave32 per edge -------------------------
__global__ void __launch_bounds__(256)
k_edge_scores(const float* __restrict__ nodef, const float* __restrict__ edgef,
              const float* __restrict__ attn_W, const float* __restrict__ attn_b,
              const int* __restrict__ dst, float* __restrict__ scores_mean,
              unsigned* __restrict__ max_enc) {
  __shared__ float sWm[NODE_DIM + EDGE_DIM];  // head-mean of attn_W
  int tid = threadIdx.x;
  if (tid < NODE_DIM + EDGE_DIM) {
    const float* r = attn_W + tid * 4;
    sWm[tid] = 0.25f * (r[0] + r[1] + r[2] + r[3]);
  }
  __syncthreads();
  int lane = tid & 31;
  int e = blockIdx.x * 8 + (tid >> 5);        // E divisible by 8
  int d = dst[e];
  const float* h  = nodef + (size_t)d * NODE_DIM;
  const float* ef = edgef + (size_t)e * EDGE_DIM;
  float p = 0.f;
#pragma unroll
  for (int i = 0; i < 4; ++i) p += h[lane + 32 * i] * sWm[lane + 32 * i];
#pragma unroll
  for (int i = 0; i < 2; ++i) p += ef[lane + 32 * i] * sWm[NODE_DIM + lane + 32 * i];
#pragma unroll
  for (int off = 16; off > 0; off >>= 1) p += __shfl_xor(p, off, 32);
  if (lane == 0) {
    p += 0.25f * (attn_b[0] + attn_b[1] + attn_b[2] + attn_b[3]);
    scores_mean[e] = p;
    atomicMax(&max_enc[d], enc_f32(p));
  }
}

// ---------------- exp + segment sum ----------------------------------------
__global__ void k_edge_exp(const float* __restrict__ scores_mean,
                           const int* __restrict__ dst,
                           const unsigned* __restrict__ max_enc,
                           float* __restrict__ exp_scores,
                           float* __restrict__ sum_exp) {
  int e = blockIdx.x * blockDim.x + threadIdx.x;
  if (e >= N_EDGES) return;
  int d = dst[e];
  float ex = __expf(scores_mean[e] - dec_f32(max_enc[d]));
  exp_scores[e] = ex;
  atomicAdd(&sum_exp[d], ex);
}

// ---------------- edge_values GEMM (WMMA f32) + weighted scatter -----------
// 64 edges per block; 8 waves = 8 column tiles of 16; 4 row-tile accumulators
// per wave sharing B fragments. A=16x4 f32 layout: lanes0-15 K=kk,kk+1;
// lanes16-31 K=kk+2,kk+3. B mirrored (K-major). C/D: rows r and r+8.
__global__ void __launch_bounds__(256)
k_edge_gemm_scatter(const float* __restrict__ edgef, const float* __restrict__ evW,
                    const float* __restrict__ evb, const int* __restrict__ dst,
                    const float* __restrict__ exp_scores,
                    const float* __restrict__ sum_exp,
                    float* __restrict__ messages) {
  __shared__ float sW[EDGE_DIM * NODE_DIM];     // 32 KB, ev_W (K-major 64x128)
  __shared__ float sA[64][EDGE_DIM + 4];        // padded rows -> conflict-free
  __shared__ float sw[64];
  __shared__ int   sd[64];
  int tid = threadIdx.x;
  int e0 = blockIdx.x * 64;                     // E divisible by 64
  for (int i = tid; i < EDGE_DIM * NODE_DIM; i += 256) sW[i] = evW[i];
  for (int i = tid; i < 64 * EDGE_DIM; i += 256) {
    int m = i >> 6, k = i & 63;
    sA[m][k] = edgef[(size_t)(e0 + m) * EDGE_DIM + k];
  }
  if (tid < 64) {
    int e = e0 + tid, d = dst[e];
    sd[tid] = d;
    sw[tid] = exp_scores[e] / (sum_exp[d] + EPS);
  }
  __syncthreads();
  int lane = tid & 31;
  int c    = tid >> 5;                          // column tile 0..7
  int mrow = lane & 15;
  int kh   = (lane >> 4) << 1;                  // 0 or 2
  int ncol = c * 16 + mrow;                     // B/C column index = lane&15
  v8f acc0 = {}, acc1 = {}, acc2 = {}, acc3 = {};
  for (int kk = 0; kk < EDGE_DIM; kk += 4) {
    v2f b; b.x = sW[(kk + kh) * NODE_DIM + ncol];
           b.y = sW[(kk + kh + 1) * NODE_DIM + ncol];
    v2f a;
    a.x = sA[mrow][kk + kh];      a.y = sA[mrow][kk + kh + 1];
    acc0 = __builtin_amdgcn_wmma_f32_16x16x4_f32(false, a, false, b, (short)0, acc0, false, false);
    a.x = sA[16 + mrow][kk + kh]; a.y = sA[16 + mrow][kk + kh + 1];
    acc1 = __builtin_amdgcn_wmma_f32_16x16x4_f32(false, a, false, b, (short)0, acc1, false, false);
    a.x = sA[32 + mrow][kk + kh]; a.y = sA[32 + mrow][kk + kh + 1];
    acc2 = __builtin_amdgcn_wmma_f32_16x16x4_f32(false, a, false, b, (short)0, acc2, false, false);
    a.x = sA[48 + mrow][kk + kh]; a.y = sA[48 + mrow][kk + kh + 1];
    acc3 = __builtin_amdgcn_wmma_f32_16x16x4_f32(false, a, false, b, (short)0, acc3, false, false);
  }
  float bn = evb[ncol];
  int mh = (lane >> 4) << 3;                    // C rows: r (+8 for upper lanes)
#pragma unroll
  for (int r = 0; r < 8; ++r) {
    int m = r + mh;
    atomicAdd(&messages[(size_t)sd[m]      * NODE_DIM + ncol], (acc0[r] + bn) * sw[m]);
    atomicAdd(&messages[(size_t)sd[16 + m] * NODE_DIM + ncol], (acc1[r] + bn) * sw[16 + m]);
    atomicAdd(&messages[(size_t)sd[32 + m] * NODE_DIM + ncol], (acc2[r] + bn) * sw[32 + m]);
    atomicAdd(&messages[(size_t)sd[48 + m] * NODE_DIM + ncol], (acc3[r] + bn) * sw[48 + m]);
  }
}

// ---------------- node update: relu(M@W1+b1)@W2+b2 + x (WMMA f32) ----------
// 32 nodes/block; W1/W2 streamed from global (L2-resident, B frags shared by
// both row tiles); t staged through LDS between the two GEMMs.
__global__ void __launch_bounds__(256)
k_node_update(const float* __restrict__ nodef, const float* __restrict__ messages,
              const float* __restrict__ W1, const float* __restrict__ b1,
              const float* __restrict__ W2, const float* __restrict__ b2,
              float* __restrict__ out) {
  __shared__ float sM[32][NODE_DIM + 4];        // messages tile, reused for t
  int tid = threadIdx.x;
  int n0 = blockIdx.x * 32;                     // N divisible by 32
  for (int i = tid; i < 32 * NODE_DIM; i += 256) {
    int m = i >> 7, k = i & 127;
    sM[m][k] = messages[(size_t)(n0 + m) * NODE_DIM + k];
  }
  __syncthreads();
  int lane = tid & 31;
  int c    = tid >> 5;
  int mrow = lane & 15;
  int kh   = (lane >> 4) << 1;
  int ncol = c * 16 + mrow;
  int mh   = (lane >> 4) << 3;

  v8f acc0 = {}, acc1 = {};
  for (int kk = 0; kk < NODE_DIM; kk += 4) {
    v2f b; b.x = W1[(size_t)(kk + kh) * NODE_DIM + ncol];
           b.y = W1[(size_t)(kk + kh + 1) * NODE_DIM + ncol];
    v2f a;
    a.x = sM[mrow][kk + kh];      a.y = sM[mrow][kk + kh + 1];
    acc0 = __builtin_amdgcn_wmma_f32_16x16x4_f32(false, a, false, b, (short)0, acc0, false, false);
    a.x = sM[16 + mrow][kk + kh]; a.y = sM[16 + mrow][kk + kh + 1];
    acc1 = __builtin_amdgcn_wmma_f32_16x16x4_f32(false, a, false, b, (short)0, acc1, false, false);
  }
  float bb1 = b1[ncol];
  __syncthreads();                              // all stage-1 reads of sM done
#pragma unroll
  for (int r = 0; r < 8; ++r) {
    int m = r + mh;
    sM[m][ncol]      = fmaxf(acc0[r] + bb1, 0.f);
    sM[16 + m][ncol] = fmaxf(acc1[r] + bb1, 0.f);
  }
  __syncthreads();

  v8f u0 = {}, u1 = {};
  for (int kk = 0; kk < NODE_DIM; kk += 4) {
    v2f b; b.x = W2[(size_t)(kk + kh) * NODE_DIM + ncol];
           b.y = W2[(size_t)(kk + kh + 1) * NODE_DIM + ncol];
    v2f a;
    a.x = sM[mrow][kk + kh];      a.y = sM[mrow][kk + kh + 1];
    u0 = __builtin_amdgcn_wmma_f32_16x16x4_f32(false, a, false, b, (short)0, u0, false, false);
    a.x = sM[16 + mrow][kk + kh]; a.y = sM[16 + mrow][kk + kh + 1];
    u1 = __builtin_amdgcn_wmma_f32_16x16x4_f32(false, a, false, b, (short)0, u1, false, false);
  }
  float bb2 = b2[ncol];
#pragma unroll
  for (int r = 0; r < 8; ++r) {
    int m = r + mh;
    size_t i0 = (size_t)(n0 + m) * NODE_DIM + ncol;
    out[i0] = nodef[i0] + u0[r] + bb2;
    size_t i1 = (size_t)(n0 + 16 + m) * NODE_DIM + ncol;
    out[i1] = nodef[i1] + u1[r] + bb2;
  }
}

extern "C" void kernel_launch(void* const* d_in, const int* in_sizes, int n_in,
                              void* d_out, int out_size, void* d_ws, size_t ws_size,
                              hipStream_t stream) {
  (void)in_sizes; (void)n_in; (void)out_size; (void)ws_size;
  const float* nodef  = (const float*)d_in[0];
  const float* edgef  = (const float*)d_in[1];
  const float* attn_W = (const float*)d_in[2];
  const float* attn_b = (const float*)d_in[3];
  const float* evW    = (const float*)d_in[4];
  const float* evb    = (const float*)d_in[5];
  const float* W1     = (const float*)d_in[6];
  const float* b1     = (const float*)d_in[7];
  const float* W2     = (const float*)d_in[8];
  const float* b2     = (const float*)d_in[9];
  const int*   eidx   = (const int*)d_in[10];
  const int*   dst    = eidx + N_EDGES;         // edge_index[1]
  float* out = (float*)d_out;

  char* ws = (char*)d_ws;
  float*    scores_mean = (float*)ws;     ws += sizeof(float) * N_EDGES;
  float*    exp_scores  = (float*)ws;     ws += sizeof(float) * N_EDGES;
  float*    sum_exp     = (float*)ws;     ws += sizeof(float) * N_NODES;
  unsigned* max_enc     = (unsigned*)ws;  ws += sizeof(unsigned) * N_NODES;
  float*    messages    = (float*)ws;     // N_NODES * NODE_DIM floats

  k_init<<<(N_NODES * NODE_DIM + 255) / 256, 256, 0, stream>>>(messages, sum_exp, max_enc);
  k_edge_scores<<<N_EDGES / 8, 256, 0, stream>>>(nodef, edgef, attn_W, attn_b, dst,
                                                 scores_mean, max_enc);
  k_edge_exp<<<(N_EDGES + 255) / 256, 256, 0, stream>>>(scores_mean, dst, max_enc,
                                                        exp_scores, sum_exp);
  k_edge_gemm_scatter<<<N_EDGES / 64, 256, 0, stream>>>(edgef, evW, evb, dst,
                                                        exp_scores, sum_exp, messages);
  k_node_update<<<N_NODES / 32, 256, 0, stream>>>(nodef, messages, W1, b1, W2, b2, out);
}